// HyperAdaGCN_69106023793434
// MI455X (gfx1250) — compile-verified
//
#include <hip/hip_runtime.h>
#include <cmath>

// ---------------------------------------------------------------------------
// HyperAdaGCN forward, gather-first formulation for MI455X (gfx1250, wave32).
// Only the NIDX indexed rows are computed. All matmuls use
// V_WMMA_F32_16X16X4_F32 (full fp32, 16x16 tiles, K stepped by 4).
// 32 rows per block; each wave computes TWO 16-row M-tiles per B fragment so
// every weight element loaded from L2 feeds two WMMAs (halves weight traffic).
// Gather staging uses GLOBAL_LOAD_ASYNC_TO_LDS_B128 when available.
// ---------------------------------------------------------------------------

typedef float v2f __attribute__((ext_vector_type(2)));
typedef float v8f __attribute__((ext_vector_type(8)));
typedef int   v4i __attribute__((ext_vector_type(4)));

typedef __attribute__((address_space(1))) v4i* glb_b128_t;   // global 128-bit
typedef __attribute__((address_space(3))) v4i* lds_b128_t;   // LDS    128-bit

#define NFEAT  512
#define NHID   128
#define NCLASS 70
#define ROWS   32    // rows (gathered samples) per block
#define LDS_S  516   // staging row stride (pad 4 -> bank-conflict-free)
#define LDS_H  132   // hidden row stride  (pad 4 -> bank-conflict-free)
#define LDS_L  72    // logits row stride

__device__ __forceinline__ v8f vzero8()
{
    v8f c = {0.f, 0.f, 0.f, 0.f, 0.f, 0.f, 0.f, 0.f};
    return c;
}

// Dual 16x16 fp32 WMMA accumulation chain sharing one B stream.
//   ar0/ar1 : per-lane A pointers for M-tile 0 / 1 (LDS, row stride lda baked in)
//   br      : per-lane B pointer (row-major weight row; B = W^T fragment)
//   lanes 0-15 supply K=k,k+1 ; lanes 16-31 supply K=k+2,k+3 (kh pre-added)
//   bscale zeroes out-of-range B columns without divergence.
__device__ __forceinline__ void wmma_chain2_f32(const float* __restrict__ ar0,
                                                const float* __restrict__ ar1,
                                                const float* __restrict__ br,
                                                float bscale, int K,
                                                v8f& c0, v8f& c1)
{
#pragma unroll 4
    for (int k = 0; k < K; k += 4) {
        v2f b, a0, a1;
        b.x  = br[k] * bscale;
        b.y  = br[k + 1] * bscale;
        a0.x = ar0[k];
        a0.y = ar0[k + 1];
        a1.x = ar1[k];
        a1.y = ar1[k + 1];
        // 8 args: (neg_a, A, neg_b, B, c_mod, C, reuse_a, reuse_b)
        c0 = __builtin_amdgcn_wmma_f32_16x16x4_f32(false, a0, false, b,
                                                   (short)0, c0, false, false);
        c1 = __builtin_amdgcn_wmma_f32_16x16x4_f32(false, a1, false, b,
                                                   (short)0, c1, false, false);
    }
}

// Gather ROWS rows (idx[row0..]) of a [N, NFEAT] matrix into LDS, then compute
// 1/L2-norm per row (0 for an all-zero row, matching the reference).
// 4 threads cooperate per row; each moves 128 floats as 32 x b128 chunks.
__device__ __forceinline__ void stage_rows(const float* __restrict__ src,
                                           const int* __restrict__ idx,
                                           int row0, int nidx,
                                           float* __restrict__ S,
                                           float (*part)[4],
                                           float* __restrict__ rinv)
{
    const int tid = threadIdx.x;
    const int r   = tid >> 2;   // 0..31
    const int sub = tid & 3;
    const int gi  = idx[min(row0 + r, nidx - 1)];
    const float* __restrict__ g = src + (size_t)gi * NFEAT + sub * 4;
    float* __restrict__ l = S + r * LDS_S + sub * 4;

#if defined(__gfx1250__) && __has_builtin(__builtin_amdgcn_global_load_async_to_lds_b128)
    // CDNA5 async copy: VGPR-bypassing global -> LDS, tracked by ASYNCcnt.
#pragma unroll
    for (int j = 0; j < 32; ++j) {
        __builtin_amdgcn_global_load_async_to_lds_b128(
            (glb_b128_t)(g + 16 * j),
            (lds_b128_t)(l + 16 * j),
            /*imm offset*/ 0, /*cpol*/ 0);
    }
#if __has_builtin(__builtin_amdgcn_s_wait_asynccnt)
    __builtin_amdgcn_s_wait_asynccnt(0);
#else
    asm volatile("s_wait_asynccnt 0" ::: "memory");
#endif
#else
    // Fallback: b128 loads through VGPRs.
#pragma unroll
    for (int j = 0; j < 32; ++j) {
        ((float4*)l)[4 * j] = ((const float4*)g)[4 * j];
    }
#endif
    __syncthreads();

    // Row sum-of-squares read back from LDS (shared by both staging paths).
    float acc = 0.f;
#pragma unroll
    for (int j = 0; j < 32; ++j) {
        float4 v = ((const float4*)l)[4 * j];
        acc += v.x * v.x + v.y * v.y + v.z * v.z + v.w * v.w;
    }
    part[r][sub] = acc;
    __syncthreads();
    if (tid < ROWS) {
        const float s = part[tid][0] + part[tid][1] + part[tid][2] + part[tid][3];
        rinv[tid] = (s > 0.f) ? (1.0f / sqrtf(s)) : 0.f;
    }
    __syncthreads();
}

__global__ __launch_bounds__(128)
void hyperadagcn_kernel(const float* __restrict__ x,
                        const float* __restrict__ sfeat,
                        const int*   __restrict__ idx,
                        const float* __restrict__ W0,
                        const float* __restrict__ W1,
                        const float* __restrict__ G1,
                        const float* __restrict__ G2,
                        const float* __restrict__ GT,
                        const float* __restrict__ Wc,
                        float* __restrict__ out,
                        int nidx)
{
    __shared__ __align__(16) float pool[ROWS * LDS_S + 2 * ROWS * LDS_H];
    __shared__ float part[ROWS][4];
    __shared__ float rinv[ROWS];
    __shared__ float rmax[ROWS];
    __shared__ float rlse[ROWS];

    const int tid  = threadIdx.x;
    const int lane = tid & 31;
    const int wave = tid >> 5;          // 4 waves; each owns 2 N-tiles x 2 M-tiles
    const int row0 = blockIdx.x * ROWS;

    float* S   = pool;                         // 32 x 512 staging (x, then sf)
    float* H0s = pool + ROWS * LDS_S;          // 32 x 128
    float* H1s = H0s + ROWS * LDS_H;           // 32 x 128
    float* G1s = pool;                         // staging region recycled
    float* G2s = G1s + ROWS * LDS_H;
    float* Fs  = G2s + ROWS * LDS_H;
    float* Ls  = Fs  + ROWS * LDS_H;           // 32 x 72 logits

    // fragment decode (32-bit 16x16 C/D layout):
    //   element c[r] sits at row = r + (lane>=16 ? 8 : 0), col = lane&15
    const int mlo = (lane >> 4) << 3;   // 0 or 8
    const int ncl = lane & 15;
    const int kh  = (lane >> 4) << 1;   // 0 or 2 (K sub-offset for A/B frags)

    // ---------------- layer 0 : H0 = relu(x @ W0^T) / ||x|| ----------------
    stage_rows(x, idx, row0, nidx, S, part, rinv);
    for (int t = 0; t < 2; ++t) {
        const int n0 = (wave * 2 + t) * 16;
        v8f c0 = vzero8(), c1 = vzero8();
        wmma_chain2_f32(S + ncl * LDS_S + kh,
                        S + (16 + ncl) * LDS_S + kh,
                        W0 + (size_t)(n0 + ncl) * NFEAT + kh,
                        1.0f, NFEAT, c0, c1);
#pragma unroll
        for (int r = 0; r < 8; ++r) {
            const int mr = r + mlo;
            H0s[mr * LDS_H + n0 + ncl]        = fmaxf(c0[r], 0.f) * rinv[mr];
            H0s[(mr + 16) * LDS_H + n0 + ncl] = fmaxf(c1[r], 0.f) * rinv[mr + 16];
        }
    }
    __syncthreads();  // all reads of S finished

    // ---------------- layer 1 : H1 = relu(sf @ W1^T) / ||sf|| ---------------
    stage_rows(sfeat, idx, row0, nidx, S, part, rinv);
    for (int t = 0; t < 2; ++t) {
        const int n0 = (wave * 2 + t) * 16;
        v8f c0 = vzero8(), c1 = vzero8();
        wmma_chain2_f32(S + ncl * LDS_S + kh,
                        S + (16 + ncl) * LDS_S + kh,
                        W1 + (size_t)(n0 + ncl) * NFEAT + kh,
                        1.0f, NFEAT, c0, c1);
#pragma unroll
        for (int r = 0; r < 8; ++r) {
            const int mr = r + mlo;
            H1s[mr * LDS_H + n0 + ncl]        = fmaxf(c0[r], 0.f) * rinv[mr];
            H1s[(mr + 16) * LDS_H + n0 + ncl] = fmaxf(c1[r], 0.f) * rinv[mr + 16];
        }
    }
    __syncthreads();  // S free; H0s/H1s complete

    // ---------------- gates : g1 = H0 @ G1^T, g2 = H1 @ G2^T ----------------
    for (int t = 0; t < 2; ++t) {
        const int n0 = (wave * 2 + t) * 16;
        v8f a0 = vzero8(), a1 = vzero8(), b0 = vzero8(), b1 = vzero8();
        wmma_chain2_f32(H0s + ncl * LDS_H + kh, H0s + (16 + ncl) * LDS_H + kh,
                        G1 + (size_t)(n0 + ncl) * NHID + kh, 1.0f, NHID, a0, a1);
        wmma_chain2_f32(H1s + ncl * LDS_H + kh, H1s + (16 + ncl) * LDS_H + kh,
                        G2 + (size_t)(n0 + ncl) * NHID + kh, 1.0f, NHID, b0, b1);
#pragma unroll
        for (int r = 0; r < 8; ++r) {
            const int mr = r + mlo;
            G1s[mr * LDS_H + n0 + ncl]        = a0[r];
            G1s[(mr + 16) * LDS_H + n0 + ncl] = a1[r];
            G2s[mr * LDS_H + n0 + ncl]        = b0[r];
            G2s[(mr + 16) * LDS_H + n0 + ncl] = b1[r];
        }
    }
    __syncthreads();

    // ------ gate = sigmoid([g1 g2] @ GT^T) ; F = (1-gate)*H0 + gate*H1 ------
    for (int t = 0; t < 2; ++t) {
        const int n0 = (wave * 2 + t) * 16;
        v8f c0 = vzero8(), c1 = vzero8();
        wmma_chain2_f32(G1s + ncl * LDS_H + kh, G1s + (16 + ncl) * LDS_H + kh,
                        GT + (size_t)(n0 + ncl) * (2 * NHID) + kh,
                        1.0f, NHID, c0, c1);
        wmma_chain2_f32(G2s + ncl * LDS_H + kh, G2s + (16 + ncl) * LDS_H + kh,
                        GT + (size_t)(n0 + ncl) * (2 * NHID) + NHID + kh,
                        1.0f, NHID, c0, c1);
#pragma unroll
        for (int r = 0; r < 8; ++r) {
            const int mr0 = r + mlo;
            const int mr1 = mr0 + 16;
            const float g0 = 1.0f / (1.0f + expf(-c0[r]));
            const float g1v = 1.0f / (1.0f + expf(-c1[r]));
            const float h00 = H0s[mr0 * LDS_H + n0 + ncl];
            const float h10 = H1s[mr0 * LDS_H + n0 + ncl];
            const float h01 = H0s[mr1 * LDS_H + n0 + ncl];
            const float h11 = H1s[mr1 * LDS_H + n0 + ncl];
            Fs[mr0 * LDS_H + n0 + ncl] = (1.0f - g0) * h00 + g0 * h10;
            Fs[mr1 * LDS_H + n0 + ncl] = (1.0f - g1v) * h01 + g1v * h11;
        }
    }
    __syncthreads();

    // ---------------- 1/||F|| per row ----------------
    if (tid < ROWS) {
        float s = 0.f;
        for (int j = 0; j < NHID; ++j) {
            const float v = Fs[tid * LDS_H + j];
            s += v * v;
        }
        rinv[tid] = (s > 0.f) ? (1.0f / sqrtf(s)) : 0.f;
    }
    __syncthreads();

    // ------- classifier : logits = relu(F @ Wc^T) / ||F||  (70 cols) --------
    // 5 N-tiles over 4 waves: wave0 -> {0,64}, waves1-3 -> {16,32,48}
    for (int n0 = wave * 16; n0 < 80; n0 += 64) {
        const int col   = n0 + ncl;
        const int colc  = min(col, NCLASS - 1);           // clamp for safe load
        const float msk = (col < NCLASS) ? 1.0f : 0.0f;   // zero padded B cols
        v8f c0 = vzero8(), c1 = vzero8();
        wmma_chain2_f32(Fs + ncl * LDS_H + kh, Fs + (16 + ncl) * LDS_H + kh,
                        Wc + (size_t)colc * NHID + kh, msk, NHID, c0, c1);
#pragma unroll
        for (int r = 0; r < 8; ++r) {
            const int mr = r + mlo;
            if (col < NCLASS) {
                Ls[mr * LDS_L + col]        = fmaxf(c0[r], 0.f) * rinv[mr];
                Ls[(mr + 16) * LDS_L + col] = fmaxf(c1[r], 0.f) * rinv[mr + 16];
            }
        }
    }
    __syncthreads();

    // ---------------- log_softmax over the 70 classes ----------------
    if (tid < ROWS) {
        float mx = -INFINITY;
        for (int cc = 0; cc < NCLASS; ++cc)
            mx = fmaxf(mx, Ls[tid * LDS_L + cc]);
        float se = 0.f;
        for (int cc = 0; cc < NCLASS; ++cc)
            se += expf(Ls[tid * LDS_L + cc] - mx);
        rmax[tid] = mx;
        rlse[tid] = logf(se);
    }
    __syncthreads();

    for (int e = tid; e < ROWS * NCLASS; e += 128) {
        const int r  = e / NCLASS;
        const int cc = e - r * NCLASS;
        if (row0 + r < nidx)
            out[(size_t)(row0 + r) * NCLASS + cc] =
                Ls[r * LDS_L + cc] - rmax[r] - rlse[r];
    }
}

extern "C" void kernel_launch(void* const* d_in, const int* in_sizes, int n_in,
                              void* d_out, int out_size, void* d_ws, size_t ws_size,
                              hipStream_t stream)
{
    const float* x   = (const float*)d_in[0];
    const float* sf  = (const float*)d_in[1];
    const int*   idx = (const int*)d_in[2];
    const float* W0  = (const float*)d_in[3];
    const float* W1  = (const float*)d_in[4];
    const float* G1  = (const float*)d_in[5];
    const float* G2  = (const float*)d_in[6];
    const float* GT  = (const float*)d_in[7];
    const float* Wc  = (const float*)d_in[8];
    float* out = (float*)d_out;

    const int nidx   = in_sizes[2];
    const int blocks = (nidx + ROWS - 1) / ROWS;   // 313 blocks for NIDX=10000

    hyperadagcn_kernel<<<blocks, 128, 0, stream>>>(x, sf, idx, W0, W1, G1, G2,
                                                   GT, Wc, out, nidx);
}